// AdapWingLoss_66958540145292
// MI455X (gfx1250) — compile-verified
//
#include <hip/hip_runtime.h>
#include <hip/hip_bf16.h>

// ---------------------------------------------------------------------------
// Adaptive Wing loss, sum reduction.  MI455X (gfx1250, wave32).
//
// Memory-bound: 537 MB streamed once (no reuse; > 192 MB L2) -> NT loads.
// theta/eps = 0.5 exactly, so ratio^p = exp2(-p) and ratio^(p-1) = 2*exp2(-p):
// the whole per-element computation reduces to 5 hardware transcendentals
// (v_exp_f32 / v_log_f32 via amdgcn builtins).
// Workgroup reduction uses V_WMMA_F32_16X16X4_F32 (ones-matrix row-sum trick,
// exact in fp32); block partials accumulate deterministically in fp64.
// ---------------------------------------------------------------------------

typedef float v2f __attribute__((ext_vector_type(2)));
typedef float v4f __attribute__((ext_vector_type(4)));
typedef float v8f __attribute__((ext_vector_type(8)));

#define AWL_THETA 0.5f
#define AWL_ALPHA 2.1f
#define AWL_OMEGA 14.0f
#define AWL_LN2   0.69314718055994531f

// Hardware transcendentals: v_exp_f32 (2^x) and v_log_f32 (log2 x).
__device__ __forceinline__ float awl_exp2(float x) {
    return __builtin_amdgcn_exp2f(x);
}
__device__ __forceinline__ float awl_log2(float x) {
    return __builtin_amdgcn_logf(x);
}
// log1p via hardware log2: absolute error ~1e-7, negligible vs sum ~1e8.
__device__ __forceinline__ float awl_log1p(float x) {
    return AWL_LN2 * awl_log2(1.0f + x);
}

__device__ __forceinline__ float awl_loss(float x, float t) {
    const float p    = AWL_ALPHA - t;          // in (1.1, 2.1]
    const float tp   = awl_exp2(-p);           // (theta/eps)^p = 0.5^p
    const float A    = AWL_OMEGA * p * (2.0f * tp) / (1.0f + tp);
    const float C    = AWL_THETA * A - AWL_OMEGA * awl_log1p(tp);
    const float diff = fabsf(t - x);
    // diff^p ; diff==0 -> exp2(p * -inf) = 0 (matches jnp.power(0,p)=0)
    const float dp   = awl_exp2(p * awl_log2(diff));
    const float nonlinear = AWL_OMEGA * awl_log1p(dp);
    const float linear    = A * diff - C;
    return (diff < AWL_THETA) ? nonlinear : linear;
}

__global__ __launch_bounds__(256) void awl_partial_kernel(
    const float* __restrict__ in, const float* __restrict__ tgt,
    double* __restrict__ ws, long long n) {
    const int tid = threadIdx.x;
    const long long gid    = (long long)blockIdx.x * blockDim.x + tid;
    const long long stride = (long long)gridDim.x * blockDim.x;

    const v4f* __restrict__ in4  = (const v4f*)in;
    const v4f* __restrict__ tgt4 = (const v4f*)tgt;
    const long long nvec = n >> 2;

    float partial = 0.0f;
    for (long long i = gid; i < nvec; i += stride) {
        v4f xi = __builtin_nontemporal_load(in4  + i);   // TH_NT: pure stream
        v4f ti = __builtin_nontemporal_load(tgt4 + i);
        partial += awl_loss(xi.x, ti.x);
        partial += awl_loss(xi.y, ti.y);
        partial += awl_loss(xi.z, ti.z);
        partial += awl_loss(xi.w, ti.w);
    }
    // scalar tail (n % 4 elements)
    for (long long i = (nvec << 2) + gid; i < n; i += stride)
        partial += awl_loss(in[i], tgt[i]);

    __shared__ float sbuf[256];
    sbuf[tid] = partial;
    __syncthreads();

    // Wave 0 reduces all 256 block partials with 4 chained fp32 WMMAs.
    // A(16x4) holds 64 partials (A[m][k] = sbuf[c*64 + 4m + k]); B = ones,
    // so D[m][n] = rowsum_k(A[m][k]) accumulated over c.  EXEC is all-ones
    // inside wave 0 (uniform branch), as WMMA requires.
    if (tid < 32) {
        const int m  = tid & 15;       // A row
        const int kh = tid >> 4;       // K half: lanes 16-31 hold K=2,3
        v2f bones; bones.x = 1.0f; bones.y = 1.0f;
        v8f acc = {};
#pragma unroll
        for (int c = 0; c < 4; ++c) {
            v2f a;
            a.x = sbuf[c * 64 + 4 * m + 2 * kh + 0];
            a.y = sbuf[c * 64 + 4 * m + 2 * kh + 1];
            acc = __builtin_amdgcn_wmma_f32_16x16x4_f32(
                /*neg_a=*/false, a, /*neg_b=*/false, bones,
                /*c_mod=*/(short)0, acc, /*reuse_a=*/false, /*reuse_b=*/false);
        }
        // lanes 0-15 hold rowsums m=0..7 in acc[0..7]; lanes 16-31 hold m=8..15
        float tot = acc[0] + acc[1] + acc[2] + acc[3] +
                    acc[4] + acc[5] + acc[6] + acc[7];
        tot += __shfl_xor(tot, 16);    // wave32 cross-half combine
        if (tid == 0) ws[blockIdx.x] = (double)tot;
    }
}

__global__ __launch_bounds__(256) void awl_finalize_kernel(
    const double* __restrict__ ws, int nb, float* __restrict__ out) {
    __shared__ double s[256];
    double acc = 0.0;
    for (int i = threadIdx.x; i < nb; i += 256)   // fixed order: deterministic
        acc += ws[i];
    s[threadIdx.x] = acc;
    __syncthreads();
    for (int off = 128; off > 0; off >>= 1) {
        if (threadIdx.x < off) s[threadIdx.x] += s[threadIdx.x + off];
        __syncthreads();
    }
    if (threadIdx.x == 0) out[0] = (float)s[0];
}

extern "C" void kernel_launch(void* const* d_in, const int* in_sizes, int n_in,
                              void* d_out, int out_size, void* d_ws, size_t ws_size,
                              hipStream_t stream) {
    const float* in  = (const float*)d_in[0];
    const float* tgt = (const float*)d_in[1];
    float* out = (float*)d_out;
    double* ws = (double*)d_ws;
    const long long n = (long long)in_sizes[0];

    int nblocks = 2048;                                   // ~8 blocks/WGP-class occupancy
    const size_t need = (size_t)nblocks * sizeof(double);
    if (ws_size < need) {
        nblocks = (int)(ws_size / sizeof(double));
        if (nblocks < 1) nblocks = 1;
    }

    awl_partial_kernel<<<nblocks, 256, 0, stream>>>(in, tgt, ws, n);
    awl_finalize_kernel<<<1, 256, 0, stream>>>(ws, nblocks, out);
}